// QueryAndGroup2_36670430773900
// MI455X (gfx1250) — compile-verified
//
#include <hip/hip_runtime.h>
#include <hip/hip_bf16.h>

// Problem constants (match reference)
constexpr int B_  = 8;
constexpr int N_  = 16384;
constexpr int S_  = 2048;
constexpr int C_  = 64;
constexpr int NS_ = 32;     // nsample
constexpr int K_  = 33;     // 1 (fps) + nsample
constexpr int CH_ = 70;     // 3 + 3 + C
constexpr float R2_ = 0.04f; // radius^2

typedef __attribute__((ext_vector_type(2))) float v2f;
typedef __attribute__((ext_vector_type(8))) float v8f;

// ---------------------------------------------------------------------------
// Kernel 1: ball query via V_WMMA_F32_16X16X4_F32.
// One wave handles 16 queries; each loop iteration scans a 32-point tile with
// two WMMAs:
//   A (16x4, loop invariant): row m = (-2qx, -2qy, -2qz, 1)
//   B (4x16):                 col n = ( px,   py,   pz, |p|^2)
//   C (16x16, loop invariant): C[m,n] = |q_m|^2 - r^2
//   D = A*B + C = d^2(m,n) - r^2   ->  hit iff D < 0
// Lanes 0..15 own one query each and build a 32-bit hit mask per tile
// (bit j -> index base+j), appending hits in ascending index order. Wave
// exits early once all 16 queries have 32 hits. Next tile's coordinates are
// prefetched into registers to overlap the b96 load latency with WMMA work.
// ---------------------------------------------------------------------------
template <bool FUSED>
__global__ __launch_bounds__(256)
void ballquery_wmma_kernel(const float* __restrict__ xyz,
                           const float* __restrict__ new_xyz,
                           const float* __restrict__ feat,
                           const int*   __restrict__ fps,
                           int*         __restrict__ wsIdx,
                           float*       __restrict__ out)
{
    extern __shared__ int lidx[];                 // [waves/block][16][K_]
    const int lane = threadIdx.x & 31;
    const int wib  = threadIdx.x >> 5;
    const int w    = blockIdx.x * (blockDim.x >> 5) + wib;
    const int q0   = w * 16;                      // global query base
    const int b    = q0 / S_;
    const int s0   = q0 % S_;                     // S_ % 16 == 0 -> same batch
    const int m    = lane & 15;
    const bool lo  = lane < 16;

    // Query coords for query m (duplicated in upper half-wave)
    const float* qp = new_xyz + ((size_t)b * S_ + (size_t)(s0 + m)) * 3;
    const float qx = qp[0], qy = qp[1], qz = qp[2];
    const float qn = qx * qx + qy * qy + qz * qz;

    // A matrix (16x4 f32): VGPR0 = K0(lanes0-15)/K2(lanes16-31), VGPR1 = K1/K3
    v2f A;
    A.x = lo ? (-2.0f * qx) : (-2.0f * qz);
    A.y = lo ? (-2.0f * qy) : 1.0f;

    // C matrix: lane L, VGPR i holds (M = i + 8*(L>=16), N = L%16) -> |q_M|^2 - r^2
    v8f Cm;
#pragma unroll
    for (int i = 0; i < 8; ++i) {
        const int src = i + ((lane & 16) >> 1);   // +8 for upper half-wave
        Cm[i] = __shfl(qn, src, 32) - R2_;
    }

    int* myidx = &lidx[((wib << 4) + m) * K_];    // owning lanes: 0..15
    int cnt = 0, first = 0;
    const float* xb = xyz + (size_t)b * N_ * 3;

    // Software-pipelined scan over 32-point tiles
    const float* p0 = xb + (size_t)m * 3;         // point (base + m)
    const float* p1 = p0 + 16 * 3;                // point (base + 16 + m)
    float c0x = p0[0], c0y = p0[1], c0z = p0[2];
    float c1x = p1[0], c1y = p1[1], c1z = p1[2];

    for (int base = 0; base < N_; base += 32) {
        // Prefetch next tile (overlaps with WMMA + ballot below)
        float n0x, n0y, n0z, n1x, n1y, n1z;
        const bool more = (base + 32) < N_;
        if (more) {
            const float* f0 = p0 + 32 * 3;
            const float* f1 = p1 + 32 * 3;
            n0x = f0[0]; n0y = f0[1]; n0z = f0[2];
            n1x = f1[0]; n1y = f1[1]; n1z = f1[2];
            p0 = f0; p1 = f1;
        }

        const float pn0 = c0x * c0x + c0y * c0y + c0z * c0z;
        const float pn1 = c1x * c1x + c1y * c1y + c1z * c1z;
        v2f B0, B1;
        B0.x = lo ? c0x : c0z;  B0.y = lo ? c0y : pn0;
        B1.x = lo ? c1x : c1z;  B1.y = lo ? c1y : pn1;

        v8f D0 = __builtin_amdgcn_wmma_f32_16x16x4_f32(
            false, A, false, B0, (short)0, Cm, false, false);
        v8f D1 = __builtin_amdgcn_wmma_f32_16x16x4_f32(
            false, A, false, B1, (short)0, Cm, false, false);

        // Route ballot words to owning lanes: lanes i and i+8 both take word i
        unsigned sel0 = 0u, sel1 = 0u;
#pragma unroll
        for (int i = 0; i < 8; ++i) {
            const unsigned b0 = (unsigned)__ballot(D0[i] < 0.0f);
            const unsigned b1 = (unsigned)__ballot(D1[i] < 0.0f);
            if ((lane & 7) == i) { sel0 = b0; sel1 = b1; }
        }
        const bool hi = (lane & 8) != 0;          // queries 8..15 -> M=i+8 half
        const unsigned m0 = hi ? (sel0 >> 16) : (sel0 & 0xFFFFu);
        const unsigned m1 = hi ? (sel1 >> 16) : (sel1 & 0xFFFFu);
        unsigned mym = m0 | (m1 << 16);           // bit j -> index base + j

        if (lo) {
            while (mym && cnt < NS_) {
                const int j  = __builtin_ctz(mym);
                const int id = base + j;
                if (cnt == 0) first = id;
                myidx[1 + cnt] = id;
                ++cnt;
                mym &= (mym - 1u);
            }
        }
        const unsigned notdone = (unsigned)__ballot(lo && (cnt < NS_));
        if (notdone == 0u) break;                 // wave-uniform early exit

        c0x = n0x; c0y = n0y; c0z = n0z;
        c1x = n1x; c1y = n1y; c1z = n1z;
    }

    if (lo) {
        const int fill = (cnt == 0) ? 0 : first;
        myidx[0] = fps[(size_t)b * S_ + (size_t)(s0 + lane)];
        for (int c = cnt; c < NS_; ++c) myidx[1 + c] = fill;
    }
    __syncthreads();

    if constexpr (!FUSED) {
        // Coalesced flush of this wave's 16*33 indices to workspace
        const int* src = &lidx[(wib << 4) * K_];
        int* dst = wsIdx + (size_t)q0 * K_;
        for (int e = lane; e < 16 * K_; e += 32) dst[e] = src[e];
    } else {
        // Fallback: gather in-wave (no workspace)
        for (int qi = 0; qi < 16; ++qi) {
            const int s = s0 + qi;
            const int* ids = &lidx[((wib << 4) + qi) * K_];
            const float nq0 = __shfl(qx, qi, 32);
            const float nq1 = __shfl(qy, qi, 32);
            const float nq2 = __shfl(qz, qi, 32);
            for (int e = lane; e < CH_ * K_; e += 32) {
                const int ch = e / K_;
                const int k  = e - ch * K_;
                const int id = ids[k];
                float v;
                if (ch < 6) {
                    const int cc = (ch >= 3) ? (ch - 3) : ch;
                    float c3 = xb[(size_t)id * 3 + cc];
                    if (ch >= 3) c3 -= (cc == 0 ? nq0 : (cc == 1 ? nq1 : nq2));
                    v = c3;
                } else {
                    v = feat[((size_t)b * C_ + (size_t)(ch - 6)) * N_ + id];
                }
                out[(((size_t)b * CH_ + ch) * S_ + s) * K_ + k] = v;
            }
        }
    }
}

// ---------------------------------------------------------------------------
// Kernel 2: grouping/gather. One block per (b, s) query: 70 channels x 33
// slots. xyz for the 33 indices staged in LDS; output writes near-coalesced
// (33 contiguous floats per channel row). Store-bandwidth bound (~151 MB out).
// ---------------------------------------------------------------------------
__global__ __launch_bounds__(256)
void gather_kernel(const float* __restrict__ xyz,
                   const float* __restrict__ new_xyz,
                   const float* __restrict__ feat,
                   const int*   __restrict__ wsIdx,
                   float*       __restrict__ out)
{
    const int bs = blockIdx.x;
    const int b  = bs >> 11;          // / S_
    const int s  = bs & (S_ - 1);     // % S_
    const int t  = threadIdx.x;

    __shared__ int   sidx[K_];
    __shared__ float sx[K_][3];
    __shared__ float snq[3];

    if (t < K_) {
        const int id = wsIdx[(size_t)bs * K_ + t];
        sidx[t] = id;
        const float* p = xyz + ((size_t)b * N_ + id) * 3;
        sx[t][0] = p[0]; sx[t][1] = p[1]; sx[t][2] = p[2];
    }
    if (t < 3) snq[t] = new_xyz[((size_t)b * S_ + s) * 3 + t];
    __syncthreads();

    const float* fb = feat + (size_t)b * C_ * N_;
    for (int e = t; e < CH_ * K_; e += 256) {
        const int ch = e / K_;
        const int k  = e - ch * K_;
        float v;
        if (ch < 3)      v = sx[k][ch];
        else if (ch < 6) v = sx[k][ch - 3] - snq[ch - 3];
        else             v = fb[(size_t)(ch - 6) * N_ + sidx[k]];
        __builtin_nontemporal_store(v, &out[(((size_t)b * CH_ + ch) * S_ + s) * K_ + k]);
    }
}

extern "C" void kernel_launch(void* const* d_in, const int* in_sizes, int n_in,
                              void* d_out, int out_size, void* d_ws, size_t ws_size,
                              hipStream_t stream)
{
    const float* xyz     = (const float*)d_in[0];   // (B, N, 3)
    const float* new_xyz = (const float*)d_in[1];   // (B, S, 3)
    const float* feat    = (const float*)d_in[2];   // (B, C, N)
    const int*   fps     = (const int*)  d_in[3];   // (B, S)
    float* out = (float*)d_out;                      // (B, 70, S, 33)

    const int    nwaves    = (B_ * S_) / 16;                 // 1024
    const int    blocks1   = nwaves / 8;                     // 8 waves/block
    const size_t lds1      = (size_t)8 * 16 * K_ * sizeof(int);
    const size_t idx_bytes = (size_t)B_ * S_ * K_ * sizeof(int);

    if (ws_size >= idx_bytes) {
        int* wsIdx = (int*)d_ws;
        ballquery_wmma_kernel<false><<<blocks1, 256, lds1, stream>>>(
            xyz, new_xyz, feat, fps, wsIdx, out);
        gather_kernel<<<B_ * S_, 256, 0, stream>>>(xyz, new_xyz, feat, wsIdx, out);
    } else {
        ballquery_wmma_kernel<true><<<blocks1, 256, lds1, stream>>>(
            xyz, new_xyz, feat, fps, nullptr, out);
    }
}